// PubmedGATConv_87093346828805
// MI455X (gfx1250) — compile-verified
//
#include <hip/hip_runtime.h>
#include <cmath>

#define N_NODES 100000
#define N_EDGES 1600000
#define H1 8
#define C1 8
#define H2 8
#define C2 3
#define NEG_SLOPE 0.2f

typedef __attribute__((ext_vector_type(2))) float v2f;
typedef __attribute__((ext_vector_type(8))) float v8f;

static __host__ int cdiv(int a, int b) { return (a + b - 1) / b; }

// ---------------------------------------------------------------------------
// Float atomic max via integer punning (works for all finite floats / -inf)
// ---------------------------------------------------------------------------
__device__ __forceinline__ void atomicMaxFloat(float* addr, float val) {
  if (val >= 0.0f) {
    atomicMax((int*)addr, __float_as_int(val));
  } else {
    atomicMin((unsigned int*)addr, __float_as_uint(val));
  }
}

// ---------------------------------------------------------------------------
// GEMM1: xw1[N,64] = x[N,500] @ W1[500,64] with V_WMMA_F32_16X16X4_F32.
// One wave owns a 16-row M-strip and computes all 4 N-tiles (64 cols),
// reusing the A fragment across 4 WMMAs per K-step -> x is read exactly once.
// A fragment (16x4 f32): lanes 0-15 = rows M=0..15, VGPR0/1 = K,K+1,
// lane-half = K offset 0/2. B fragment (4x16) mirrored: lane%16 = column.
// 100000 = 6250*16 and 500 = 125*4: no tails inside the wave.
// ---------------------------------------------------------------------------
__global__ __launch_bounds__(256)
void gemm1_wmma(const float* __restrict__ x, const float* __restrict__ W,
                float* __restrict__ out) {
  const int lane  = threadIdx.x & 31;
  const int gwave = (blockIdx.x * blockDim.x + threadIdx.x) >> 5;
  if (gwave >= N_NODES / 16) return;  // wave-uniform: EXEC all-ones below
  const int m0 = gwave << 4;
  const int la = lane & 15;  // row (A) / col-within-tile (B)
  const int lh = lane >> 4;  // K half selector

  const float* __restrict__ arow = x + (size_t)(m0 + la) * 500 + 2 * lh;
  const float* __restrict__ brow = W + (size_t)(2 * lh) * 64 + la;

  v8f acc0 = {}, acc1 = {}, acc2 = {}, acc3 = {};
  for (int k = 0; k < 500; k += 4) {
    v2f a;
    a.x = arow[k];
    a.y = arow[k + 1];
    const float* b0 = brow + (size_t)k * 64;
    v2f b[4];
#pragma unroll
    for (int t = 0; t < 4; ++t) {
      b[t].x = b0[t * 16];
      b[t].y = b0[64 + t * 16];
    }
    acc0 = __builtin_amdgcn_wmma_f32_16x16x4_f32(false, a, false, b[0],
                                                 (short)0, acc0, false, false);
    acc1 = __builtin_amdgcn_wmma_f32_16x16x4_f32(false, a, false, b[1],
                                                 (short)0, acc1, false, false);
    acc2 = __builtin_amdgcn_wmma_f32_16x16x4_f32(false, a, false, b[2],
                                                 (short)0, acc2, false, false);
    acc3 = __builtin_amdgcn_wmma_f32_16x16x4_f32(false, a, false, b[3],
                                                 (short)0, acc3, false, false);
  }

  // C/D layout: VGPR v, lanes 0-15 -> M=v, lanes 16-31 -> M=v+8; N = lane%16
  const int mrow = m0 + lh * 8;
  float* __restrict__ orow = out + (size_t)mrow * 64 + la;
#pragma unroll
  for (int v = 0; v < 8; ++v) {
    orow[(size_t)v * 64 + 0]  = acc0[v];
    orow[(size_t)v * 64 + 16] = acc1[v];
    orow[(size_t)v * 64 + 32] = acc2[v];
    orow[(size_t)v * 64 + 48] = acc3[v];
  }
}

// Pad W2[64,24] -> W2p[64,32] (zeros in cols 24..31)
__global__ __launch_bounds__(256)
void pad_w2_k(const float* __restrict__ W2, float* __restrict__ W2p) {
  int t = blockIdx.x * blockDim.x + threadIdx.x;
  if (t >= 64 * 32) return;
  int r = t >> 5, c = t & 31;
  W2p[t] = (c < 24) ? W2[r * 24 + c] : 0.f;
}

// ---------------------------------------------------------------------------
// GEMM2: xw2[N,24] = h[N,64] @ W2p[64,32] via WMMA; one wave per 16-row
// strip, 2 N-tiles (A reused), K = 64 = 16*4. Only cols < 24 stored.
// ---------------------------------------------------------------------------
__global__ __launch_bounds__(256)
void gemm2_wmma(const float* __restrict__ h, const float* __restrict__ W2p,
                float* __restrict__ out) {
  const int lane  = threadIdx.x & 31;
  const int gwave = (blockIdx.x * blockDim.x + threadIdx.x) >> 5;
  if (gwave >= N_NODES / 16) return;  // wave-uniform
  const int m0 = gwave << 4;
  const int la = lane & 15;
  const int lh = lane >> 4;

  const float* __restrict__ arow = h + (size_t)(m0 + la) * 64 + 2 * lh;
  const float* __restrict__ brow = W2p + (size_t)(2 * lh) * 32 + la;

  v8f acc0 = {}, acc1 = {};
#pragma unroll
  for (int k = 0; k < 64; k += 4) {
    v2f a, b0, b1;
    a.x = arow[k];
    a.y = arow[k + 1];
    const float* bp = brow + (size_t)k * 32;
    b0.x = bp[0];
    b0.y = bp[32];
    b1.x = bp[16];
    b1.y = bp[48];
    acc0 = __builtin_amdgcn_wmma_f32_16x16x4_f32(false, a, false, b0,
                                                 (short)0, acc0, false, false);
    acc1 = __builtin_amdgcn_wmma_f32_16x16x4_f32(false, a, false, b1,
                                                 (short)0, acc1, false, false);
  }

  const int mrow = m0 + lh * 8;
#pragma unroll
  for (int v = 0; v < 8; ++v) {
    out[(size_t)(mrow + v) * 24 + la] = acc0[v];          // cols 0..15
    if (la < 8)                                           // cols 16..23 only
      out[(size_t)(mrow + v) * 24 + 16 + la] = acc1[v];
  }
}

// ---------------------------------------------------------------------------
// Per-(node,head) attention logits: als[n,h] = xw[n,h,:] . a_src[h,:]
// ---------------------------------------------------------------------------
template <int H, int C>
__global__ __launch_bounds__(256)
void attn_logits_k(const float* __restrict__ xw, const float* __restrict__ a_src,
                   const float* __restrict__ a_dst, float* __restrict__ als,
                   float* __restrict__ ald, int N) {
  int t = blockIdx.x * blockDim.x + threadIdx.x;
  if (t >= N * H) return;
  int n = t / H, h = t % H;
  const float* row = xw + (size_t)n * (H * C) + h * C;
  float s0 = 0.f, s1 = 0.f;
#pragma unroll
  for (int c = 0; c < C; ++c) {
    s0 += row[c] * a_src[h * C + c];
    s1 += row[c] * a_dst[h * C + c];
  }
  als[t] = s0;
  ald[t] = s1;
}

__global__ __launch_bounds__(256)
void fill_k(float* __restrict__ p, float v, int n) {
  int t = blockIdx.x * blockDim.x + threadIdx.x;
  if (t < n) p[t] = v;
}

// ---------------------------------------------------------------------------
// Edge passes. Edge ids [0,E) come from edge_index; ids [E, E+N) are the
// implicit self loops (src=dst=id-E), matching PyG add_self_loops=True.
// ---------------------------------------------------------------------------
template <int H>
__global__ __launch_bounds__(256)
void edge_max_k(const int* __restrict__ src, const int* __restrict__ dst, int E,
                int N, const float* __restrict__ als,
                const float* __restrict__ ald, float* __restrict__ m) {
  int e = blockIdx.x * blockDim.x + threadIdx.x;
  if (e >= E + N) return;
  int s = (e < E) ? src[e] : (e - E);
  int d = (e < E) ? dst[e] : (e - E);
#pragma unroll
  for (int h = 0; h < H; ++h) {
    float v = als[s * H + h] + ald[d * H + h];
    v = (v > 0.f) ? v : NEG_SLOPE * v;
    atomicMaxFloat(&m[d * H + h], v);
  }
}

template <int H>
__global__ __launch_bounds__(256)
void edge_sum_k(const int* __restrict__ src, const int* __restrict__ dst, int E,
                int N, const float* __restrict__ als,
                const float* __restrict__ ald, const float* __restrict__ m,
                float* __restrict__ sbuf) {
  int e = blockIdx.x * blockDim.x + threadIdx.x;
  if (e >= E + N) return;
  int s = (e < E) ? src[e] : (e - E);
  int d = (e < E) ? dst[e] : (e - E);
#pragma unroll
  for (int h = 0; h < H; ++h) {
    float v = als[s * H + h] + ald[d * H + h];
    v = (v > 0.f) ? v : NEG_SLOPE * v;
    atomicAdd(&sbuf[d * H + h], expf(v - m[d * H + h]));
  }
}

template <int H, int C>
__global__ __launch_bounds__(256)
void edge_scatter_k(const int* __restrict__ src, const int* __restrict__ dst,
                    int E, int N, const float* __restrict__ als,
                    const float* __restrict__ ald, const float* __restrict__ m,
                    const float* __restrict__ sbuf, const float* __restrict__ xw,
                    float* __restrict__ acc) {
  int e = blockIdx.x * blockDim.x + threadIdx.x;
  if (e >= E + N) return;
  int s = (e < E) ? src[e] : (e - E);
  int d = (e < E) ? dst[e] : (e - E);
#pragma unroll
  for (int h = 0; h < H; ++h) {
    float v = als[s * H + h] + ald[d * H + h];
    v = (v > 0.f) ? v : NEG_SLOPE * v;
    float alpha = expf(v - m[d * H + h]) / (sbuf[d * H + h] + 1e-16f);
    const float* xr = xw + (size_t)s * (H * C) + h * C;
    float* ar = acc + (size_t)d * (H * C) + h * C;
#pragma unroll
    for (int c = 0; c < C; ++c) atomicAdd(&ar[c], alpha * xr[c]);
  }
}

// h = elu(acc1 + b1)
__global__ __launch_bounds__(256)
void finalize1_k(const float* __restrict__ acc, const float* __restrict__ b,
                 float* __restrict__ hbuf, int N) {
  int t = blockIdx.x * blockDim.x + threadIdx.x;
  if (t >= N * 64) return;
  float v = acc[t] + b[t & 63];
  hbuf[t] = (v > 0.f) ? v : (expf(v) - 1.f);
}

// mean over heads + b2, row softmax -> d_out[N,3]
__global__ __launch_bounds__(256)
void finalize2_k(const float* __restrict__ acc2, const float* __restrict__ b2,
                 float* __restrict__ out, int N) {
  int n = blockIdx.x * blockDim.x + threadIdx.x;
  if (n >= N) return;
  float o[3];
#pragma unroll
  for (int c = 0; c < 3; ++c) {
    float s = 0.f;
#pragma unroll
    for (int h = 0; h < 8; ++h) s += acc2[(size_t)n * 24 + h * 3 + c];
    o[c] = s * 0.125f + b2[c];
  }
  float mx = fmaxf(o[0], fmaxf(o[1], o[2]));
  float e0 = expf(o[0] - mx), e1 = expf(o[1] - mx), e2 = expf(o[2] - mx);
  float inv = 1.f / (e0 + e1 + e2);
  out[(size_t)n * 3 + 0] = e0 * inv;
  out[(size_t)n * 3 + 1] = e1 * inv;
  out[(size_t)n * 3 + 2] = e2 * inv;
}

// ---------------------------------------------------------------------------
extern "C" void kernel_launch(void* const* d_in, const int* in_sizes, int n_in,
                              void* d_out, int out_size, void* d_ws,
                              size_t ws_size, hipStream_t stream) {
  const float* x      = (const float*)d_in[0];
  const float* W1     = (const float*)d_in[1];
  const float* a1_src = (const float*)d_in[2];
  const float* a1_dst = (const float*)d_in[3];
  const float* b1     = (const float*)d_in[4];
  const float* W2     = (const float*)d_in[5];
  const float* a2_src = (const float*)d_in[6];
  const float* a2_dst = (const float*)d_in[7];
  const float* b2     = (const float*)d_in[8];
  const int*   ei     = (const int*)d_in[9];   // [2, E] int32 (JAX x64 off)
  const int* src = ei;
  const int* dst = ei + N_EDGES;
  float* out = (float*)d_out;

  const int N = N_NODES, E = N_EDGES, ET = E + N;

  // workspace layout (floats): ~64 MB total
  float* ws   = (float*)d_ws;
  float* xw1  = ws;                         // N*64 ; reused as h after layer 1
  float* acc1 = ws + (size_t)N * 64;        // N*64 ; reused: xw2 | acc2
  float* als  = ws + (size_t)N * 128;       // N*8
  float* ald  = als + (size_t)N * 8;        // N*8
  float* mb   = ald + (size_t)N * 8;        // N*8
  float* sb   = mb  + (size_t)N * 8;        // N*8
  float* W2p  = sb  + (size_t)N * 8;        // 64*32 padded W2
  float* hbuf = xw1;                        // alias (xw1 dead after scatter)
  float* xw2  = acc1;                       // N*24
  float* acc2 = acc1 + (size_t)N * 24;      // N*24

  const int B = 256;
  const int gEdges = cdiv(ET, B);
  const int gStrip = cdiv(N / 16, B / 32);  // one wave per 16-row strip

  // ---- layer 1 ----
  gemm1_wmma<<<gStrip, B, 0, stream>>>(x, W1, xw1);
  attn_logits_k<H1, C1><<<cdiv(N * H1, B), B, 0, stream>>>(xw1, a1_src, a1_dst,
                                                           als, ald, N);
  fill_k<<<cdiv(N * H1, B), B, 0, stream>>>(mb, -INFINITY, N * H1);
  fill_k<<<cdiv(N * H1, B), B, 0, stream>>>(sb, 0.f, N * H1);
  fill_k<<<cdiv(N * 64, B), B, 0, stream>>>(acc1, 0.f, N * 64);
  edge_max_k<H1><<<gEdges, B, 0, stream>>>(src, dst, E, N, als, ald, mb);
  edge_sum_k<H1><<<gEdges, B, 0, stream>>>(src, dst, E, N, als, ald, mb, sb);
  edge_scatter_k<H1, C1><<<gEdges, B, 0, stream>>>(src, dst, E, N, als, ald, mb,
                                                   sb, xw1, acc1);
  finalize1_k<<<cdiv(N * 64, B), B, 0, stream>>>(acc1, b1, hbuf, N);

  // ---- layer 2 ----
  pad_w2_k<<<cdiv(64 * 32, B), B, 0, stream>>>(W2, W2p);
  gemm2_wmma<<<gStrip, B, 0, stream>>>(hbuf, W2p, xw2);
  attn_logits_k<H2, C2><<<cdiv(N * H2, B), B, 0, stream>>>(xw2, a2_src, a2_dst,
                                                           als, ald, N);
  fill_k<<<cdiv(N * H2, B), B, 0, stream>>>(mb, -INFINITY, N * H2);
  fill_k<<<cdiv(N * H2, B), B, 0, stream>>>(sb, 0.f, N * H2);
  fill_k<<<cdiv(N * 24, B), B, 0, stream>>>(acc2, 0.f, N * 24);
  edge_max_k<H2><<<gEdges, B, 0, stream>>>(src, dst, E, N, als, ald, mb);
  edge_sum_k<H2><<<gEdges, B, 0, stream>>>(src, dst, E, N, als, ald, mb, sb);
  edge_scatter_k<H2, C2><<<gEdges, B, 0, stream>>>(src, dst, E, N, als, ald, mb,
                                                   sb, xw2, acc2);
  finalize2_k<<<cdiv(N, B), B, 0, stream>>>(acc2, b2, out, N);
}